// SpIN_57612691308955
// MI455X (gfx1250) — compile-verified
//
#include <hip/hip_runtime.h>
#include <hip/hip_bf16.h>
#include <math.h>
#include <stdint.h>

typedef __attribute__((ext_vector_type(2))) float v2f;
typedef __attribute__((ext_vector_type(8))) float v8f;

#define BATCH 1024
#define DIN   512
#define NE    16

// ---- workspace layout (float offsets) ----
#define WS_PHI   0        // 1024*16 = 16384
#define WS_C2    16384    // 16*512  = 8192   C = 2*phi^T @ x / B
#define WS_SIGMA 24576    // 256              sigma = phi^T phi / B
#define WS_PI    24832    // 256              pi    = phi^T Tphi / B
#define WS_SVEC  25088    // 16               s     = 2*colsum(phi)/B
#define WS_GSIG  25104    // 256              gsigma

// ---- output layout (flat concat: loss, eigvals, chol, grad_W, grad_b, sigma_new) ----
#define OUT_LOSS  0
#define OUT_EIG   1
#define OUT_CHOL  17
#define OUT_GRADW 273
#define OUT_GRADB 8465
#define OUT_SIGMA 8481

// -----------------------------------------------------------------------------
// Kernel 1: phi = x @ W^T + b  via V_WMMA_F32_16X16X4_F32
// one wave per 16-row tile of phi; K=512 stepped by 4
// A(16x4) layout: lane%16 = M row; lane/16 picks K-pair {0,1} vs {2,3}
// B(4x16)  layout: lane%16 = N col; same K-pair striping
// C/D(16x16): VGPR j holds M=j (lanes 0-15) and M=j+8 (lanes 16-31), N=lane%16
// -----------------------------------------------------------------------------
__global__ void __launch_bounds__(32)
spin_phi_wmma(const float* __restrict__ x, const float* __restrict__ W,
              const float* __restrict__ bias, float* __restrict__ ws) {
  float* phi = ws + WS_PHI;
  const int lane = threadIdx.x;
  const int m    = lane & 15;
  const int kh   = (lane >> 4) << 1;      // 0 or 2
  const int row0 = blockIdx.x << 4;

  const float* xrow = x + (size_t)(row0 + m) * DIN + kh;
  const float* wrow = W + (size_t)m * DIN + kh;

  v8f acc = {0.f,0.f,0.f,0.f,0.f,0.f,0.f,0.f};
#pragma unroll 4
  for (int k = 0; k < DIN; k += 4) {
    v2f a = *(const v2f*)(xrow + k);      // x[row][k+kh], x[row][k+kh+1]
    v2f b = *(const v2f*)(wrow + k);      // W[n][k+kh],   W[n][k+kh+1]
    acc = __builtin_amdgcn_wmma_f32_16x16x4_f32(false, a, false, b,
                                                (short)0, acc, false, false);
  }

  const float bn = bias[m];
  const int   r0 = row0 + ((lane >> 4) << 3);
#pragma unroll
  for (int j = 0; j < 8; ++j)
    phi[(size_t)(r0 + j) * NE + m] = acc[j] + bn;
}

// -----------------------------------------------------------------------------
// Kernel 2: all phi^T-reductions via WMMA, K = 1024.
// phi (64KB) is async-staged HBM/L2 -> LDS with GLOBAL_LOAD_ASYNC_TO_LDS_B128,
// then the WMMA A-operand streams from LDS (ds_load path).
//   tiles 0..31 : C2[:, t*16 .. t*16+15] = (2/B) * phi^T @ x
//   tile  32    : pi    = phi^T @ Tphi / B
//   tile  33    : sigma = phi^T @ phi  / B   (B operand == A operand here)
// -----------------------------------------------------------------------------
__global__ void __launch_bounds__(32)
spin_cov_wmma(const float* __restrict__ x, const float* __restrict__ Tphi,
              float* __restrict__ ws) {
  __shared__ float lphi[BATCH * NE];              // 64 KB of the 320 KB/WGP LDS
  const int lane = threadIdx.x;
  const int m    = lane & 15;
  const int kh   = (lane >> 4) << 1;
  const int tile = blockIdx.x;

  // ---- async stage phi -> LDS: 128 iters x 32 lanes x 16B = 64 KB ----
  {
    const float* gphi = ws + WS_PHI;
    unsigned int       lbase = (unsigned int)(uintptr_t)&lphi[0];   // low 32 bits = LDS offset
    unsigned long long gbase = (unsigned long long)(uintptr_t)gphi;
#pragma unroll 4
    for (int it = 0; it < 128; ++it) {
      unsigned int       loff  = lbase + (unsigned int)((it * 32 + lane) * 16);
      unsigned long long gaddr = gbase + (unsigned long long)((it * 32 + lane) * 16);
      asm volatile("global_load_async_to_lds_b128 %0, %1, off"
                   :: "v"(loff), "v"(gaddr) : "memory");
    }
    asm volatile("s_wait_asynccnt 0" ::: "memory");   // single-wave block: no barrier needed
  }

  const float* Bm; int bld, boff, ostride; float scale; float* outp;
  if (tile < 32) {
    Bm = x;    bld = DIN; boff = tile * 16; ostride = DIN;
    scale = 2.0f / (float)BATCH; outp = ws + WS_C2 + boff;
  } else if (tile == 32) {
    Bm = Tphi; bld = NE;  boff = 0;         ostride = NE;
    scale = 1.0f / (float)BATCH; outp = ws + WS_PI;
  } else {
    Bm = Tphi; bld = NE;  boff = 0;         ostride = NE;   // Bm unused for tile 33
    scale = 1.0f / (float)BATCH; outp = ws + WS_SIGMA;
  }

  v8f acc = {0.f,0.f,0.f,0.f,0.f,0.f,0.f,0.f};
#pragma unroll 2
  for (int k = 0; k < BATCH; k += 4) {
    v2f a, b;
    a.x = lphi[(k + kh)     * NE + m];            // A = phi^T : A[m][k]  (LDS)
    a.y = lphi[(k + kh + 1) * NE + m];
    if (tile < 33) {
      b.x = Bm[(size_t)(k + kh)     * bld + boff + m];
      b.y = Bm[(size_t)(k + kh + 1) * bld + boff + m];
    } else {
      b = a;                                      // sigma: B[k][n] = phi[k][n], n == m
    }
    acc = __builtin_amdgcn_wmma_f32_16x16x4_f32(false, a, false, b,
                                                (short)0, acc, false, false);
  }

  const int mr0 = (lane >> 4) << 3;
#pragma unroll
  for (int j = 0; j < 8; ++j)
    outp[(size_t)(mr0 + j) * ostride + m] = acc[j] * scale;

  if (tile == 33 && lane < 16) {                  // svec = 2*colsum(phi)/B  (from LDS)
    float s = 0.f;
    for (int k = 0; k < BATCH; ++k) s += lphi[k * NE + lane];
    ws[WS_SVEC + lane] = s * (2.0f / (float)BATCH);
  }
}

// -----------------------------------------------------------------------------
// Kernel 3: tiny 16x16 serial chain (Cholesky, inverse, Lambda, gsigma, grad_b)
// ~10 KFLOP total: single thread, no point parallelizing.
// -----------------------------------------------------------------------------
__global__ void __launch_bounds__(32)
spin_small(const float* __restrict__ sigma_avg, const float* __restrict__ j_avg_b,
           float* __restrict__ ws, float* __restrict__ out) {
  if (threadIdx.x != 0) return;
  const float* sig_raw = ws + WS_SIGMA;
  const float* pi      = ws + WS_PI;
  const float* svec    = ws + WS_SVEC;

  float S[NE][NE], L[NE][NE], Li[NE][NE], TMP[NE][NE], Lam[NE][NE], T[NE][NE], G[NE][NE];

  // sigma_new = 0.99*sigma_avg + 0.01*sigma; output it; add 1e-3*I for chol
  for (int i = 0; i < NE; ++i)
    for (int j = 0; j < NE; ++j) {
      float v = 0.99f * sigma_avg[i*NE+j] + 0.01f * sig_raw[i*NE+j];
      out[OUT_SIGMA + i*NE + j] = v;
      S[i][j] = v + ((i == j) ? 1e-3f : 0.f);
    }

  // Cholesky (lower), zero upper
  for (int j = 0; j < NE; ++j) {
    float d = S[j][j];
    for (int k = 0; k < j; ++k) d = fmaf(-L[j][k], L[j][k], d);
    float lj = sqrtf(d);
    L[j][j] = lj;
    for (int i = j + 1; i < NE; ++i) {
      float v = S[i][j];
      for (int k = 0; k < j; ++k) v = fmaf(-L[i][k], L[j][k], v);
      L[i][j] = v / lj;
    }
    for (int i = 0; i < j; ++i) L[i][j] = 0.f;
  }

  // Li = inv(L), lower triangular
  for (int i = 0; i < NE; ++i)
    for (int j = 0; j < NE; ++j) Li[i][j] = 0.f;
  for (int j = 0; j < NE; ++j) {
    Li[j][j] = 1.0f / L[j][j];
    for (int i = j + 1; i < NE; ++i) {
      float v = 0.f;
      for (int k = j; k < i; ++k) v = fmaf(L[i][k], Li[k][j], v);
      Li[i][j] = -v / L[i][i];
    }
  }

  // Lambda = Li @ pi @ Li^T
  for (int i = 0; i < NE; ++i)
    for (int j = 0; j < NE; ++j) {
      float v = 0.f;
      for (int k = 0; k < NE; ++k) v = fmaf(Li[i][k], pi[k*NE+j], v);
      TMP[i][j] = v;
    }
  for (int i = 0; i < NE; ++i)
    for (int j = 0; j < NE; ++j) {
      float v = 0.f;
      for (int k = 0; k < NE; ++k) v = fmaf(TMP[i][k], Li[j][k], v);
      Lam[i][j] = v;
    }

  // loss, eigvals, chol out
  float loss = 0.f;
  for (int i = 0; i < NE; ++i) { out[OUT_EIG + i] = Lam[i][i]; loss += Lam[i][i]; }
  out[OUT_LOSS] = loss;
  for (int i = 0; i < NE; ++i)
    for (int j = 0; j < NE; ++j) out[OUT_CHOL + i*NE + j] = L[i][j];

  // T = triu(Lam @ diag(diag(Li))); G = Li^T @ T
  for (int i = 0; i < NE; ++i)
    for (int j = 0; j < NE; ++j)
      T[i][j] = (j >= i) ? Lam[i][j] * Li[j][j] : 0.f;
  for (int i = 0; i < NE; ++i)
    for (int j = 0; j < NE; ++j) {
      float v = 0.f;
      for (int k = 0; k < NE; ++k) v = fmaf(Li[k][i], T[k][j], v);
      G[i][j] = v;
      ws[WS_GSIG + i*NE + j] = v;
    }

  // grad_b[n] = 0.99*sum_{l,m} G[l][m]*j_avg_b[l,m,n] + 0.01*sum_l G[l][n]*svec[l]
  for (int n = 0; n < NE; ++n) {
    float a1 = 0.f;
    for (int l = 0; l < NE; ++l)
      for (int mm = 0; mm < NE; ++mm)
        a1 = fmaf(G[l][mm], j_avg_b[(size_t)(l*NE + mm)*NE + n], a1);
    float a2 = 0.f;
    for (int l = 0; l < NE; ++l) a2 = fmaf(G[l][n], svec[l], a2);
    out[OUT_GRADB + n] = 0.99f * a1 + 0.01f * a2;
  }
}

// -----------------------------------------------------------------------------
// Kernel 4: grad_W — streams the 8 MB j_avg_W once with b128 loads (512B per
// wave32 issue slot), fully coalesced; gsigma is wave-uniform -> scalar loads.
// Each thread owns 4 consecutive d's (4 | 512, so n is uniform per thread).
// -----------------------------------------------------------------------------
__global__ void __launch_bounds__(256)
spin_gradw(const float* __restrict__ jW, const float* __restrict__ ws,
           float* __restrict__ out) {
  const float* gsig = ws + WS_GSIG;
  const float* C2   = ws + WS_C2;
  const int t    = blockIdx.x * blockDim.x + threadIdx.x;   // 0..2047
  const int base = t << 2;                                  // 0..8188
  const int n    = base >> 9;
  const int d    = base & 511;

  float4 acc = make_float4(0.f, 0.f, 0.f, 0.f);
#pragma unroll 4
  for (int lm = 0; lm < 256; ++lm) {
    int p = (lm + 8 < 256) ? (lm + 8) : 255;
    __builtin_prefetch(jW + (size_t)p * 8192 + base);
    const float  g = gsig[lm];
    const float4 v = *reinterpret_cast<const float4*>(jW + (size_t)lm * 8192 + base);
    acc.x = fmaf(g, v.x, acc.x);
    acc.y = fmaf(g, v.y, acc.y);
    acc.z = fmaf(g, v.z, acc.z);
    acc.w = fmaf(g, v.w, acc.w);
  }

  float4 acc2 = make_float4(0.f, 0.f, 0.f, 0.f);
#pragma unroll
  for (int l = 0; l < NE; ++l) {
    const float  g = gsig[l*NE + n];
    const float4 c = *reinterpret_cast<const float4*>(C2 + (size_t)l * DIN + d);
    acc2.x = fmaf(g, c.x, acc2.x);
    acc2.y = fmaf(g, c.y, acc2.y);
    acc2.z = fmaf(g, c.z, acc2.z);
    acc2.w = fmaf(g, c.w, acc2.w);
  }

  out[OUT_GRADW + base + 0] = 0.99f * acc.x + 0.01f * acc2.x;
  out[OUT_GRADW + base + 1] = 0.99f * acc.y + 0.01f * acc2.y;
  out[OUT_GRADW + base + 2] = 0.99f * acc.z + 0.01f * acc2.z;
  out[OUT_GRADW + base + 3] = 0.99f * acc.w + 0.01f * acc2.w;
}

// -----------------------------------------------------------------------------
extern "C" void kernel_launch(void* const* d_in, const int* in_sizes, int n_in,
                              void* d_out, int out_size, void* d_ws, size_t ws_size,
                              hipStream_t stream) {
  (void)in_sizes; (void)n_in; (void)out_size; (void)ws_size;
  const float* x         = (const float*)d_in[0];   // [1024,512]
  const float* W         = (const float*)d_in[1];   // [16,512]
  const float* b         = (const float*)d_in[2];   // [16]
  const float* Tphi      = (const float*)d_in[3];   // [1024,16]
  const float* sigma_avg = (const float*)d_in[4];   // [16,16]
  const float* j_avg_W   = (const float*)d_in[5];   // [16,16,16,512]
  const float* j_avg_b   = (const float*)d_in[6];   // [16,16,16]
  float* out = (float*)d_out;
  float* ws  = (float*)d_ws;

  spin_phi_wmma<<<64, 32, 0, stream>>>(x, W, b, ws);
  spin_cov_wmma<<<34, 32, 0, stream>>>(x, Tphi, ws);
  spin_small  <<< 1, 32, 0, stream>>>(sigma_avg, j_avg_b, ws, out);
  spin_gradw  <<< 8, 256, 0, stream>>>(j_avg_W, ws, out);
}